// SlotSelector_81690277970459
// MI455X (gfx1250) — compile-verified
//
#include <hip/hip_runtime.h>
#include <hip/hip_bf16.h>

#define NKEYS 32768
#define DH    256
#define KSLOT 128
#define TTOP  512
#define MMEM  8
#define TOPP  640      // 512 + 128 headroom: masked top-512 always subset of unmasked top-640
#define CAP   2048

typedef float v2f __attribute__((ext_vector_type(2)));
typedef float v8f __attribute__((ext_vector_type(8)));

// ---- monotonic float <-> uint32 key (descending value == descending key) ----
__device__ __forceinline__ unsigned fkey(float f) {
    unsigned u = __float_as_uint(f);
    return (u & 0x80000000u) ? ~u : (u | 0x80000000u);
}
__device__ __forceinline__ float funkey(unsigned k) {
    unsigned u = (k & 0x80000000u) ? (k & 0x7FFFFFFFu) : ~k;
    return __uint_as_float(u);
}

// ---- in-LDS bitonic ascending sort of n (power of 2) 64-bit composites ----
__device__ void bitonic_sort_asc(unsigned long long* a, int n, int tid, int nthr) {
    for (int k2 = 2; k2 <= n; k2 <<= 1) {
        for (int j = k2 >> 1; j > 0; j >>= 1) {
            for (int i = tid; i < n; i += nthr) {
                int ixj = i ^ j;
                if (ixj > i) {
                    unsigned long long x = a[i], y = a[ixj];
                    bool asc = ((i & k2) == 0);
                    if (asc ? (x > y) : (x < y)) { a[i] = y; a[ixj] = x; }
                }
            }
            __syncthreads();
        }
    }
}

// ---------------- l2norm rows (DH = 256, one block per row) ----------------
__global__ __launch_bounds__(256) void l2norm_rows(const float* __restrict__ in,
                                                   float* __restrict__ out) {
    __shared__ float red[256];
    int r = blockIdx.x, t = threadIdx.x;
    float x = in[(size_t)r * DH + t];
    red[t] = x * x;
    __syncthreads();
    for (int off = 128; off >= 1; off >>= 1) {
        if (t < off) red[t] += red[t + off];
        __syncthreads();
    }
    float denom = sqrtf(red[0]) + 1e-6f;
    out[(size_t)r * DH + t] = x / denom;
}

// --------- logits = slot_n @ keys_n^T via V_WMMA_F32_16X16X4_F32 -----------
// one wave = one 16x16 tile; waves in a block share the keys (B) tile column.
__global__ __launch_bounds__(256) void wmma_logits(const float* __restrict__ slot_n,
                                                   const float* __restrict__ keys_n,
                                                   float* __restrict__ logits) {
    int lane = threadIdx.x & 31;
    int wave = threadIdx.x >> 5;
    int tile = blockIdx.x * 8 + wave;     // 16384 tiles total
    int tm = tile & 7;                    // 8 M-tiles (128/16)
    int tn = tile >> 3;                   // 2048 N-tiles (32768/16)
    int m0 = tm * 16, n0 = tn * 16;
    int r = lane & 15;
    int koff = (lane >> 4) * 2;           // lanes 16-31 hold K=2,3

    // A 16x4 f32: lane(0-15): v0=A[r][0],v1=A[r][1]; lane(16-31): v0=A[r][2],v1=A[r][3]
    // B 4x16 f32: lane(0-15): v0=B[0][r],v1=B[1][r]; lane(16-31): v0=B[2][r],v1=B[3][r]
    const v2f* ap = (const v2f*)(slot_n + (size_t)(m0 + r) * DH + koff);
    const v2f* bp = (const v2f*)(keys_n + (size_t)(n0 + r) * DH + koff);

    v8f c = {};
#pragma unroll 4
    for (int kk = 0; kk < DH / 4; ++kk) {
        v2f a = ap[2 * kk];
        v2f b = bp[2 * kk];
        c = __builtin_amdgcn_wmma_f32_16x16x4_f32(false, a, false, b,
                                                  (short)0, c, false, false);
    }
    // C/D: VGPR v, lanes 0-15 -> M=v, lanes 16-31 -> M=v+8, N=lane%16
    int col = n0 + r;
    int rbase = m0 + ((lane >> 4) ? 8 : 0);
#pragma unroll
    for (int v = 0; v < 8; ++v)
        logits[(size_t)(rbase + v) * NKEYS + col] = c[v];
}

// ------------------------- zero-fill (float4) ------------------------------
__global__ void zero_f4(float4* __restrict__ p, size_t n4) {
    size_t i = (size_t)blockIdx.x * blockDim.x + threadIdx.x;
    if (i < n4) p[i] = make_float4(0.f, 0.f, 0.f, 0.f);
}

// --- per-row exact top-TOPP (value desc, index asc) via radix-select -------
__global__ __launch_bounds__(1024) void topk_rows(const float* __restrict__ logits,
                                                  unsigned long long* __restrict__ cand_out) {
    __shared__ unsigned hist[256];
    __shared__ unsigned long long cand[CAP];
    __shared__ int s_cnt;
    __shared__ unsigned s_hi;
    __shared__ int s_rem;

    int k = blockIdx.x;
    int tid = threadIdx.x;
    const float* row = logits + (size_t)k * NKEYS;

    unsigned hi = 0;
    int rem = TOPP;
    for (int pass = 0; pass < 4; ++pass) {
        int shift = 24 - 8 * pass;
        if (tid < 256) hist[tid] = 0u;
        __syncthreads();
        for (int i = tid; i < NKEYS; i += 1024) {
            unsigned key = fkey(row[i]);
            bool ok = (pass == 0) || ((key >> (shift + 8)) == hi);
            if (ok) atomicAdd(&hist[(key >> shift) & 255u], 1u);
        }
        __syncthreads();
        if (tid == 0) {
            unsigned cum = 0;
            int v = 0;
            for (int b = 255; b >= 0; --b) {
                unsigned h = hist[b];
                if (cum + h >= (unsigned)rem) { v = b; break; }
                cum += h;
            }
            s_hi = (hi << 8) | (unsigned)v;
            s_rem = rem - (int)cum;
        }
        __syncthreads();
        hi = s_hi;
        rem = s_rem;
        __syncthreads();
    }
    unsigned thr = hi;   // exact TOPP-th key; (#keys > thr) = TOPP - rem < TOPP

    if (tid == 0) s_cnt = 0;
    __syncthreads();
    for (int i = tid; i < NKEYS; i += 1024) {
        unsigned key = fkey(row[i]);
        if (key >= thr) {
            int p = atomicAdd(&s_cnt, 1);
            if (p < CAP)
                cand[p] = ((unsigned long long)key << 32) | (unsigned)(~(unsigned)i);
        }
    }
    __syncthreads();
    int cnt = s_cnt < CAP ? s_cnt : CAP;
    for (int i = cnt + tid; i < CAP; i += 1024) cand[i] = 0ull;
    __syncthreads();
    bitonic_sort_asc(cand, CAP, tid, 1024);   // composite order == top_k order
    for (int t = tid; t < TOPP; t += 1024)
        cand_out[(size_t)k * TOPP + t] = cand[CAP - 1 - t];
}

// ------------- sequential 128-step scan (single workgroup) -----------------
__global__ __launch_bounds__(1024) void scan_steps(
    const unsigned long long* __restrict__ cand_in,  // [K][TOPP]
    const float* __restrict__ gumbel,                // [K][T]
    const float* __restrict__ tau_p,
    const float* __restrict__ keys_n,                // [N][DH]
    const float* __restrict__ mem_cache,             // [N][M*DH]
    float* __restrict__ out_mem,                     // [K][M*DH]
    float* __restrict__ out_idx,                     // [K]
    float* __restrict__ out_qp,                      // [K][N]
    float* __restrict__ out_qst,                     // [K][N]
    float* __restrict__ out_ek)                      // [K][DH]
{
    __shared__ unsigned mask[NKEYS / 32];            // 4 KB
    __shared__ unsigned long long comp[1024];        // 8 KB
    __shared__ float s_s[TTOP];
    __shared__ float s_soft[TTOP];
    __shared__ int   s_idx[TTOP];
    __shared__ float s_red[TTOP];
    __shared__ unsigned long long s_am[TTOP];        // argmax composites
    __shared__ float s_acc[4][DH];

    int tid = threadIdx.x;
    float tau = tau_p[0];
    for (int i = tid; i < NKEYS / 32; i += 1024) mask[i] = 0u;
    __syncthreads();

    for (int k = 0; k < KSLOT; ++k) {
        // gather precomputed candidates, drop masked ones
        unsigned long long c = (tid < TOPP) ? cand_in[(size_t)k * TOPP + tid] : 0ull;
        if (c != 0ull) {
            unsigned idx = ~(unsigned)(c & 0xFFFFFFFFu);
            if (mask[idx >> 5] & (1u << (idx & 31))) c = 0ull;
        }
        comp[tid] = c;
        __syncthreads();
        bitonic_sort_asc(comp, 1024, tid, 1024);

        // top-512 in top_k order: (value desc, index asc)
        if (tid < TTOP) {
            unsigned long long cc = comp[1023 - tid];
            s_idx[tid] = (int)(~(unsigned)(cc & 0xFFFFFFFFu));
            float v = funkey((unsigned)(cc >> 32));
            float s = (v + gumbel[(size_t)k * TTOP + tid]) / tau;
            s_s[tid] = s;
            s_red[tid] = s;
        }
        __syncthreads();
        // max reduce
        for (int off = 256; off >= 1; off >>= 1) {
            if (tid < off) { float a = s_red[tid], b = s_red[tid + off]; s_red[tid] = a > b ? a : b; }
            __syncthreads();
        }
        float smax = s_red[0];
        __syncthreads();
        if (tid < TTOP) { float e = expf(s_s[tid] - smax); s_soft[tid] = e; s_red[tid] = e; }
        __syncthreads();
        for (int off = 256; off >= 1; off >>= 1) {
            if (tid < off) s_red[tid] += s_red[tid + off];
            __syncthreads();
        }
        float esum = s_red[0];
        __syncthreads();
        if (tid < TTOP) {
            float sv = s_soft[tid] / esum;
            s_soft[tid] = sv;
            // argmax(soft), first-occurrence tie-break (matches jnp.argmax)
            s_am[tid] = ((unsigned long long)fkey(sv) << 32) | (unsigned)(~(unsigned)tid);
        }
        __syncthreads();
        for (int off = 256; off >= 1; off >>= 1) {
            if (tid < off) { unsigned long long x = s_am[tid], y = s_am[tid + off]; if (y > x) s_am[tid] = y; }
            __syncthreads();
        }
        int hidx = (int)(~(unsigned)(s_am[0] & 0xFFFFFFFFu));
        int idx_hard = s_idx[hidx];
        float sh = s_soft[hidx];
        float wh = (1.0f - sh) + sh;          // straight-through weight at hard slot

        // scatter q_probs / q_st (buffers pre-zeroed; w_t == 0 exactly for t != hidx)
        if (tid < TTOP) {
            out_qp [(size_t)k * NKEYS + s_idx[tid]] = s_soft[tid];
            out_qst[(size_t)k * NKEYS + s_idx[tid]] = (tid == hidx) ? wh : 0.0f;
        }
        if (tid == 0) out_idx[k] = (float)idx_hard;

        // Mem_sel[k] = wh * mem_cache[idx_hard]  (exact: w one-hot except hard slot)
        for (int j = tid; j < MMEM * DH; j += 1024)
            out_mem[(size_t)k * (MMEM * DH) + j] = wh * mem_cache[(size_t)idx_hard * (MMEM * DH) + j];

        // exp_key = soft @ keys_n[topi]
        {
            int g = tid >> 8, d = tid & 255;
            float acc = 0.f;
            for (int t = g * 128; t < g * 128 + 128; ++t)
                acc += s_soft[t] * keys_n[(size_t)s_idx[t] * DH + d];
            s_acc[g][d] = acc;
        }
        __syncthreads();
        if (tid < DH)
            out_ek[(size_t)k * DH + tid] =
                ((s_acc[0][tid] + s_acc[1][tid]) + s_acc[2][tid]) + s_acc[3][tid];

        if (tid == 0) mask[idx_hard >> 5] |= (1u << (idx_hard & 31));
        __syncthreads();
    }
}

extern "C" void kernel_launch(void* const* d_in, const int* in_sizes, int n_in,
                              void* d_out, int out_size, void* d_ws, size_t ws_size,
                              hipStream_t stream) {
    const float* keys      = (const float*)d_in[0];   // [32768,256]
    const float* mem_cache = (const float*)d_in[1];   // [32768,8,256]
    /* d_in[2] y_cache: unused by reference */
    const float* slot_q    = (const float*)d_in[3];   // [128,256]
    const float* gumbel    = (const float*)d_in[4];   // [128,512]
    const float* tau       = (const float*)d_in[5];   // scalar

    float* out      = (float*)d_out;
    float* out_mem  = out;                                   // 128*8*256   = 262144
    float* out_idx  = out + 262144;                          // 128
    float* out_log  = out + 262272;                          // 128*32768   = 4194304
    float* out_qp   = out + 4456576;                         // 4194304
    float* out_qst  = out + 8650880;                         // 4194304
    float* out_ek   = out + 12845184;                        // 128*256     = 32768

    char* ws = (char*)d_ws;
    float* keys_n = (float*)ws;                              // 32 MB
    float* slot_n = (float*)(ws + (size_t)NKEYS * DH * 4);   // 128 KB
    unsigned long long* cand =
        (unsigned long long*)(ws + (size_t)NKEYS * DH * 4 + (size_t)KSLOT * DH * 4);

    l2norm_rows<<<NKEYS, 256, 0, stream>>>(keys, keys_n);
    l2norm_rows<<<KSLOT, 256, 0, stream>>>(slot_q, slot_n);
    wmma_logits<<<2048, 256, 0, stream>>>(slot_n, keys_n, out_log);
    zero_f4<<<2048, 1024, 0, stream>>>((float4*)out_qp, (size_t)2097152); // q_probs+q_st
    topk_rows<<<KSLOT, 1024, 0, stream>>>(out_log, cand);
    scan_steps<<<1, 1024, 0, stream>>>(cand, gumbel, tau, keys_n, mem_cache,
                                       out_mem, out_idx, out_qp, out_qst, out_ek);
}